// GCNLayer_19524921327986
// MI455X (gfx1250) — compile-verified
//
#include <hip/hip_runtime.h>

#define N_NODES 100000
#define N_EDGES 1600000
#define D 64

typedef float v2f __attribute__((ext_vector_type(2)));
typedef float v8f __attribute__((ext_vector_type(8)));

// ---------------------------------------------------------------------------
// Kernel 1: zero workspace (sum[N*D] followed by deg[N]); total divisible by 4.
// ---------------------------------------------------------------------------
__global__ void gcn_zero_ws_kernel(float4* __restrict__ ws, int n4) {
    int i = blockIdx.x * blockDim.x + threadIdx.x;
    if (i < n4) ws[i] = make_float4(0.f, 0.f, 0.f, 0.f);
}

// ---------------------------------------------------------------------------
// Kernel 2: edge scatter. One wave (32 lanes) per edge; each lane moves a
// float2 of the 64-float source row (coalesced 256B read, L2-resident) and
// atomically accumulates into the destination row. Lane 0 counts degree.
// ---------------------------------------------------------------------------
__global__ void __launch_bounds__(256)
gcn_scatter_kernel(const float* __restrict__ h,
                   const int* __restrict__ esrc,
                   const int* __restrict__ edst,
                   float* __restrict__ sum,
                   float* __restrict__ deg) {
    unsigned gid  = blockIdx.x * blockDim.x + threadIdx.x;
    unsigned e    = gid >> 5;      // wave-uniform edge id
    unsigned lane = gid & 31u;
    if (e >= N_EDGES) return;

    int s = esrc[e];
    int t = edst[e];

    const float2* hp = (const float2*)(h + (size_t)s * D);
    float2 v = hp[lane];

    float* sp = sum + (size_t)t * D + lane * 2;
    atomicAdd(sp,     v.x);
    atomicAdd(sp + 1, v.y);
    if (lane == 0) atomicAdd(deg + t, 1.0f);
}

// ---------------------------------------------------------------------------
// Kernel 3: out = mask(deg>0) * relu( (sum * 1/max(deg,1)) @ W^T + b )
// One wave per 16x16 output tile using V_WMMA_F32_16X16X4_F32, K-loop of 16.
// Grid sized exactly (25000 tiles / 8 waves per block = 3125 blocks) so there
// is no divergent early-exit: EXEC is all-ones at every WMMA (ISA requirement).
//
// A (16x4 f32, 2 VGPR): lane L holds row M = L%16; K = kk + 2*(L/16) + {0,1}.
// B (4x16 f32, 2 VGPR): lane L holds col N = L%16; same K split. B[k][n]=W[n][k].
// C/D (16x16 f32, 8 VGPR): VGPR r -> M = r + 8*(L/16), N = L%16.
// ---------------------------------------------------------------------------
__global__ void __launch_bounds__(256)
gcn_gemm_relu_kernel(const float* __restrict__ sum,
                     const float* __restrict__ deg,
                     const float* __restrict__ Wmat,
                     const float* __restrict__ bvec,
                     float* __restrict__ out) {
    int wave = (blockIdx.x * blockDim.x + threadIdx.x) >> 5;   // tile id
    int lane = threadIdx.x & 31;
    int tm   = wave >> 2;          // row-tile   (0..6249)
    int tn   = wave & 3;           // col-tile   (0..3)
    int m0   = tm << 4;
    int n0   = tn << 4;
    int lm   = lane & 15;          // row (A) / col (B,C,D) within tile
    int lh   = lane >> 4;          // half-wave selector

    // Per-row mean scale for the A operand (row = m0 + lm for this lane).
    float rdeg = 1.0f / fmaxf(deg[m0 + lm], 1.0f);

    const float* arow = sum  + (size_t)(m0 + lm) * D;   // aggregated features
    const float* wrow = Wmat + (size_t)(n0 + lm) * D;   // W[n][*] (B = W^T)

    v8f acc = {};
#pragma unroll
    for (int kk = 0; kk < D; kk += 4) {
        int ka = kk + (lh << 1);
        v2f a, bm;
        a.x  = arow[ka]     * rdeg;
        a.y  = arow[ka + 1] * rdeg;
        bm.x = wrow[ka];
        bm.y = wrow[ka + 1];
        acc = __builtin_amdgcn_wmma_f32_16x16x4_f32(
                  /*neg_a=*/false, a, /*neg_b=*/false, bm,
                  /*c_mod=*/(short)0, acc, /*reuse_a=*/false, /*reuse_b=*/false);
    }

    float bias = bvec[n0 + lm];
#pragma unroll
    for (int r = 0; r < 8; ++r) {
        int m = m0 + r + (lh << 3);
        float val = fmaxf(acc[r] + bias, 0.0f);
        if (deg[m] == 0.0f) val = 0.0f;   // zero-in-degree nodes stay 0 (DGL mean)
        out[(size_t)m * D + n0 + lm] = val;
    }
}

// ---------------------------------------------------------------------------
extern "C" void kernel_launch(void* const* d_in, const int* in_sizes, int n_in,
                              void* d_out, int out_size, void* d_ws, size_t ws_size,
                              hipStream_t stream) {
    const float* h    = (const float*)d_in[0];
    const int*   esrc = (const int*)d_in[1];
    const int*   edst = (const int*)d_in[2];
    const float* Wmat = (const float*)d_in[3];
    const float* bvec = (const float*)d_in[4];
    float*       out  = (float*)d_out;

    float* sum = (float*)d_ws;                       // [N_NODES * D]
    float* deg = sum + (size_t)N_NODES * D;          // [N_NODES]

    // 1) zero sums + degrees (26 MB, divisible by 16 bytes)
    int n4 = (N_NODES * D + N_NODES) / 4;
    gcn_zero_ws_kernel<<<(n4 + 255) / 256, 256, 0, stream>>>((float4*)d_ws, n4);

    // 2) edge scatter: one wave per edge
    long long threads = (long long)N_EDGES * 32;
    gcn_scatter_kernel<<<(int)((threads + 255) / 256), 256, 0, stream>>>(
        h, esrc, edst, sum, deg);

    // 3) scaled GEMM + bias + relu + zero-degree mask via WMMA
    int tiles  = (N_NODES / 16) * 4;   // 25000, divisible by 8 waves/block
    int blocks = tiles / 8;            // 3125
    gcn_gemm_relu_kernel<<<blocks, 256, 0, stream>>>(sum, deg, Wmat, bvec, out);
}